// GraphSageLayer_49916109914435
// MI455X (gfx1250) — compile-verified
//
#include <hip/hip_runtime.h>
#include <hip/hip_bf16.h>

#define DD 128

typedef float v2f __attribute__((ext_vector_type(2)));
typedef float v8f __attribute__((ext_vector_type(8)));

// ---------------------------------------------------------------------------
// Kernel 1: zero the scratch (agg[N*128] followed by deg[N], contiguous)
// ---------------------------------------------------------------------------
__global__ void __launch_bounds__(256) sage_zero_kernel(float* __restrict__ p, int count) {
    int i = blockIdx.x * blockDim.x + threadIdx.x;
    if (i < count) p[i] = 0.0f;
}

// ---------------------------------------------------------------------------
// Kernel 2: edge scatter-add.  One 32-lane wave per edge:
//   - coalesced float4 gather of x[src] (global_load_b128)
//   - 4 non-returning agent-scope f32 atomics into agg[dst]
//   - lane 0 counts degree
// x and agg both fit in the 192MB L2, so these atomics stay on-chip.
// ---------------------------------------------------------------------------
__global__ void __launch_bounds__(256) sage_scatter_kernel(
    const float* __restrict__ x, const int* __restrict__ ei,
    float* __restrict__ agg, float* __restrict__ deg, int nE)
{
    int t    = blockIdx.x * blockDim.x + threadIdx.x;
    int e    = t >> 5;
    int lane = t & 31;
    if (e >= nE) return;

    int src = ei[e];
    int dst = ei[nE + e];

    const float4* xs = (const float4*)(x + (size_t)src * DD);
    float4 v = xs[lane];

    float* a = agg + (size_t)dst * DD + lane * 4;
    __hip_atomic_fetch_add(a + 0, v.x, __ATOMIC_RELAXED, __HIP_MEMORY_SCOPE_AGENT);
    __hip_atomic_fetch_add(a + 1, v.y, __ATOMIC_RELAXED, __HIP_MEMORY_SCOPE_AGENT);
    __hip_atomic_fetch_add(a + 2, v.z, __ATOMIC_RELAXED, __HIP_MEMORY_SCOPE_AGENT);
    __hip_atomic_fetch_add(a + 3, v.w, __ATOMIC_RELAXED, __HIP_MEMORY_SCOPE_AGENT);
    if (lane == 0)
        __hip_atomic_fetch_add(deg + dst, 1.0f, __ATOMIC_RELAXED, __HIP_MEMORY_SCOPE_AGENT);
}

// ---------------------------------------------------------------------------
// Kernel 3: fused  out = relu(x @ Ws^T + (agg/deg) @ Wn^T + bs + bn)
// One wave computes a 16x128 output tile with V_WMMA_F32_16X16X4_F32.
//
// f32 16x16x4 fragment layouts (wave32):
//   A (16x4):  lane m=lane&15 holds row M=m; VGPR0/1 hold K=(lane>>4)*2, +1
//              -> contiguous float2 from row-major X / AGG
//   B (4x16):  lane holds col N=lane&15; VGPR0/1 hold K=(lane>>4)*2, +1
//              -> y = x@W^T means B[k][n] = W[n][k]: contiguous float2 from
//                 row-major W.
//   C/D (16x16): VGPR v -> row r0 + v + (lane>>4)*8, col = lane&15.
// deg normalization is folded into the neigh A-fragment (invd per row).
// ---------------------------------------------------------------------------
__global__ void __launch_bounds__(256) sage_gemm_kernel(
    const float* __restrict__ x,  const float* __restrict__ Ws,
    const float* __restrict__ bs, const float* __restrict__ Wn,
    const float* __restrict__ bn, const float* __restrict__ agg,
    const float* __restrict__ deg, float* __restrict__ out, int n)
{
    int wave = (blockIdx.x * blockDim.x + threadIdx.x) >> 5;
    int lane = threadIdx.x & 31;
    int r0   = wave * 16;
    if (r0 >= n) return;            // whole wave exits together: EXEC stays all-1s

    int m     = lane & 15;
    int khalf = lane >> 4;          // 0 or 1
    int row   = r0 + m;

    float invd = 1.0f / fmaxf(deg[row], 1.0f);

    const float* xrow = x   + (size_t)row * DD;
    const float* arow = agg + (size_t)row * DD;

    v8f acc[8] = {};                // 8 column tiles of 16 -> full 128 cols

    for (int k0 = 0; k0 < DD; k0 += 4) {
        int ka = k0 + khalf * 2;
        v2f aX = *(const v2f*)(xrow + ka);
        v2f aN;
        aN.x = arow[ka]     * invd;
        aN.y = arow[ka + 1] * invd;
#pragma unroll
        for (int nt = 0; nt < 8; nt++) {
            int col = nt * 16 + m;
            v2f bS = *(const v2f*)(Ws + (size_t)col * DD + ka);
            v2f bN = *(const v2f*)(Wn + (size_t)col * DD + ka);
            acc[nt] = __builtin_amdgcn_wmma_f32_16x16x4_f32(
                false, aX, false, bS, (short)0, acc[nt], false, false);
            acc[nt] = __builtin_amdgcn_wmma_f32_16x16x4_f32(
                false, aN, false, bN, (short)0, acc[nt], false, false);
        }
    }

#pragma unroll
    for (int nt = 0; nt < 8; nt++) {
        int col  = nt * 16 + m;
        float bias = bs[col] + bn[col];
#pragma unroll
        for (int v = 0; v < 8; v++) {
            int rr = r0 + v + khalf * 8;
            float val = acc[nt][v] + bias;
            out[(size_t)rr * DD + col] = fmaxf(val, 0.0f);
        }
    }
}

// ---------------------------------------------------------------------------
extern "C" void kernel_launch(void* const* d_in, const int* in_sizes, int n_in,
                              void* d_out, int out_size, void* d_ws, size_t ws_size,
                              hipStream_t stream) {
    const float* x  = (const float*)d_in[0];
    const int*   ei = (const int*)  d_in[1];
    const float* Ws = (const float*)d_in[2];
    const float* bs = (const float*)d_in[3];
    const float* Wn = (const float*)d_in[4];
    const float* bn = (const float*)d_in[5];
    float* out = (float*)d_out;

    const int n  = in_sizes[0] / DD;   // 50000
    const int nE = in_sizes[1] / 2;    // 800000

    float* agg = (float*)d_ws;                       // N*128 floats
    float* deg = agg + (size_t)n * DD;               // N floats

    // 1) zero scratch (harness does not re-poison between replays)
    int zc = n * DD + n;
    sage_zero_kernel<<<(zc + 255) / 256, 256, 0, stream>>>(agg, zc);

    // 2) scatter-sum + degree: one wave per edge
    long long sthreads = (long long)nE * 32;
    sage_scatter_kernel<<<(unsigned)((sthreads + 255) / 256), 256, 0, stream>>>(
        x, ei, agg, deg, nE);

    // 3) fused WMMA GEMM + bias + relu: one wave per 16 output rows
    long long gwaves  = (n + 15) / 16;                // 3125
    long long gthreads = gwaves * 32;
    sage_gemm_kernel<<<(unsigned)((gthreads + 255) / 256), 256, 0, stream>>>(
        x, Ws, bs, Wn, bn, agg, deg, out, n);
}